// TopoBrainGPU_18769007084219
// MI455X (gfx1250) — compile-verified
//
#include <hip/hip_runtime.h>
#include <math.h>

// ---------------------------------------------------------------------------
// TopoBrain pipeline for MI455X (gfx1250), fp32 WMMA path.
// B=512 F=512 G=32 N=1024 D=64 H=64 C=100 A=8
// ---------------------------------------------------------------------------

typedef float v2f __attribute__((ext_vector_type(2)));
typedef float v8f __attribute__((ext_vector_type(8)));

#define BATCH 512
#define FEAT  512
#define GRID  32
#define NN    1024
#define DD    64
#define HH    64
#define CC    100
#define AA    8
#define ND    65536          // N*D
#define BN    524288         // B*N
#define NTILES 32768         // BN/16
#define TS    65             // padded LDS tile stride (bank-conflict free)
#define R1CHUNK 2048         // split-K chunk for readout GEMM1
#define R1NCH   32           // 65536 / 2048

__device__ __forceinline__ float sigmoidf_(float x) { return 1.0f / (1.0f + expf(-x)); }
__device__ __forceinline__ float gelu_erf(float x) {
  return 0.5f * x * (1.0f + erff(x * 0.70710678118654752440f));
}

__device__ __forceinline__ v8f wmma4(v2f a, v2f b, v8f c) {
  return __builtin_amdgcn_wmma_f32_16x16x4_f32(false, a, false, b, (short)0, c,
                                               false, false);
}

// One wave computes one 16x16 f32 tile: C += A[16xK] * B[Kx16].
__device__ __forceinline__ v8f wmma_k(const float* __restrict__ A, int lda,
                                      const float* __restrict__ B, int ldb,
                                      int K, v8f c) {
  const int lane  = threadIdx.x & 31;
  const int mrow  = lane & 15;
  const int khalf = lane >> 4;
  for (int k = 0; k < K; k += 4) {
    const int kk = k + 2 * khalf;
    v2f a, b;
    a.x = A[(size_t)mrow * lda + kk];
    a.y = A[(size_t)mrow * lda + kk + 1];
    b.x = B[(size_t)kk * ldb + mrow];
    b.y = B[(size_t)(kk + 1) * ldb + mrow];
    c = wmma4(a, b, c);
  }
  return c;
}

// ---------------------------------------------------------------------------
// K0: Kb = basis @ Wk  [8,64];  ortho scalar -> out_ortho
// ---------------------------------------------------------------------------
__global__ __launch_bounds__(256) void k_prep(const float* __restrict__ basis,
                                              const float* __restrict__ Wk,
                                              float* __restrict__ Kb,
                                              float* __restrict__ out_ortho) {
  __shared__ float sBas[AA * DD];
  __shared__ float sRed[64];
  const int tid = threadIdx.x;
  for (int i = tid; i < AA * DD; i += 256) sBas[i] = basis[i];
  __syncthreads();
  for (int o = tid; o < AA * DD; o += 256) {
    const int a = o >> 6, d = o & 63;
    float acc = 0.f;
    for (int k = 0; k < DD; ++k) acc += sBas[a * DD + k] * Wk[k * DD + d];
    Kb[o] = acc;
  }
  if (tid < 64) {
    const int a1 = tid >> 3, a2 = tid & 7;
    float g = 0.f;
    for (int k = 0; k < DD; ++k) g += sBas[a1 * DD + k] * sBas[a2 * DD + k];
    const float dv = g - (a1 == a2 ? 1.f : 0.f);
    sRed[tid] = dv * dv;
  }
  __syncthreads();
  if (tid == 0) {
    float s = 0.f;
    for (int i = 0; i < 64; ++i) s += sRed[i];
    *out_ortho = fminf(fmaxf(s, 0.f), 10.f);
  }
}

// ---------------------------------------------------------------------------
// K2: per-node normalized 4-neighbor adjacency weights (up,down,left,right).
// Row sum taken honestly over the full masked row (matches reference deg).
// ---------------------------------------------------------------------------
__global__ __launch_bounds__(256) void k_adj(const float* __restrict__ aw,
                                             const float* __restrict__ am,
                                             float* __restrict__ adj4) {
  const int wave = threadIdx.x >> 5, lane = threadIdx.x & 31;
  const int node = blockIdx.x * 8 + wave;
  const float* wr = aw + (size_t)node * NN;
  const float* mr = am + (size_t)node * NN;
  float s = 0.f;
  for (int j = lane; j < NN; j += 32) s += sigmoidf_(wr[j]) * mr[j];
  for (int off = 16; off > 0; off >>= 1) s += __shfl_down(s, off);
  if (lane == 0) {
    const float inv = 1.0f / fmaxf(s, 1e-6f);
    const int r = node >> 5, c = node & 31;
    float4 o;
    o.x = (r > 0)        ? sigmoidf_(wr[node - GRID]) * mr[node - GRID] * inv : 0.f;
    o.y = (r < GRID - 1) ? sigmoidf_(wr[node + GRID]) * mr[node + GRID] * inv : 0.f;
    o.z = (c > 0)        ? sigmoidf_(wr[node - 1])    * mr[node - 1]    * inv : 0.f;
    o.w = (c < GRID - 1) ? sigmoidf_(wr[node + 1])    * mr[node + 1]    * inv : 0.f;
    reinterpret_cast<float4*>(adj4)[node] = o;
  }
}

// ---------------------------------------------------------------------------
// K1: x_embed = x @ W_embed + b_embed   [512x512]x[512x65536]
// One wave computes a 16(M) x 64(N) strip: 4 accumulators, A reused 4x.
// ---------------------------------------------------------------------------
__global__ __launch_bounds__(256) void k_gemm_embed(const float* __restrict__ X,
                                                    const float* __restrict__ W,
                                                    const float* __restrict__ bias,
                                                    float* __restrict__ C) {
  const int wave = threadIdx.x >> 5;
  const int t = blockIdx.x * 8 + wave;            // 32768 wave strips
  const int mt = t >> 10;                         // 0..31
  const int nb = t & 1023;                        // 0..1023
  const int m0 = mt * 16, n0 = nb * 64;
  const int lane = threadIdx.x & 31;
  const int mrow = lane & 15, khalf = lane >> 4;

  const float* Arow = X + (size_t)(m0 + mrow) * FEAT;
  const float* Bp   = W + n0 + mrow;
  v8f c0 = {}, c1 = {}, c2 = {}, c3 = {};
  for (int k = 0; k < FEAT; k += 4) {
    const int kk = k + 2 * khalf;
    v2f a, b;
    a.x = Arow[kk];
    a.y = Arow[kk + 1];
    const float* B0 = Bp + (size_t)kk * ND;
    const float* B1 = Bp + (size_t)(kk + 1) * ND;
    b.x = B0[0];  b.y = B1[0];  c0 = wmma4(a, b, c0);
    b.x = B0[16]; b.y = B1[16]; c1 = wmma4(a, b, c1);
    b.x = B0[32]; b.y = B1[32]; c2 = wmma4(a, b, c2);
    b.x = B0[48]; b.y = B1[48]; c3 = wmma4(a, b, c3);
  }
  v8f acc[4] = {c0, c1, c2, c3};
  for (int j = 0; j < 4; ++j) {
    const int n = n0 + j * 16 + mrow;
    const float bn = bias[n];
    for (int v = 0; v < 8; ++v) {
      const int m = m0 + v + 8 * khalf;
      C[(size_t)m * ND + n] = acc[j][v] + bn;
    }
  }
}

// ---------------------------------------------------------------------------
// K3: fused sparse diffusion + node MLP + Q-proj + basis softmax + x_clean.
// Block = 256 threads = 8 waves, wave handles a 16-row tile of [B*N, 64].
// ---------------------------------------------------------------------------
__global__ __launch_bounds__(256) void k_node(
    const float* __restrict__ xemb, const float* __restrict__ adj4,
    const float* __restrict__ W1, const float* __restrict__ b1,
    const float* __restrict__ W2, const float* __restrict__ b2,
    const float* __restrict__ Wq, const float* __restrict__ Kb_g,
    const float* __restrict__ basis_g, float* __restrict__ xflat,
    float* __restrict__ entpart) {
  __shared__ float sW1[DD * DD], sW2[DD * DD], sWq[DD * DD];
  __shared__ float sb1[DD], sb2[DD];
  __shared__ float sKb[AA * DD], sBas[AA * DD];
  __shared__ float sA[8][16 * TS];
  __shared__ float sB[8][16 * TS];
  __shared__ float sAt[8][16 * AA];

  const int tid = threadIdx.x;
  for (int i = tid; i < DD * DD; i += 256) {
    sW1[i] = W1[i]; sW2[i] = W2[i]; sWq[i] = Wq[i];
  }
  for (int i = tid; i < AA * DD; i += 256) { sKb[i] = Kb_g[i]; sBas[i] = basis_g[i]; }
  if (tid < DD) { sb1[tid] = b1[tid]; sb2[tid] = b2[tid]; }
  __syncthreads();

  const int wave = tid >> 5, lane = tid & 31;
  const int t = blockIdx.x * 8 + wave;              // tile id, < 32768
  const int r0 = t * 16;                            // flat row start
  const int bidx = r0 >> 10;                        // batch
  const int i0 = r0 & (NN - 1);                     // node start
  const int mrow = lane & 15, khalf = lane >> 4;

  // ---- stage 1: sparse degree-normalized aggregation -> sA ----
  const float* xb = xemb + (size_t)bidx * NN * DD;
  for (int it = 0; it < 32; ++it) {
    const int v = it * 32 + lane;
    const int row = v >> 6, d = v & 63;
    const int node = i0 + row;
    const float4 w4 = reinterpret_cast<const float4*>(adj4)[node];
    const int up = node - GRID < 0 ? 0 : node - GRID;
    const int dn = node + GRID > NN - 1 ? NN - 1 : node + GRID;
    const int lf = node - 1 < 0 ? 0 : node - 1;
    const int rt = node + 1 > NN - 1 ? NN - 1 : node + 1;
    const float acc = w4.x * xb[up * DD + d] + w4.y * xb[dn * DD + d] +
                      w4.z * xb[lf * DD + d] + w4.w * xb[rt * DD + d];
    sA[wave][row * TS + d] = acc;
  }
  __syncthreads();

  // ---- stage 2: h = gelu(sA @ W1 + b1) -> sB ----
  for (int nt = 0; nt < 4; ++nt) {
    v8f c = {};
    c = wmma_k(&sA[wave][0], TS, &sW1[nt * 16], DD, DD, c);
    const int n = nt * 16 + mrow;
    const float bn = sb1[n];
    for (int v = 0; v < 8; ++v) {
      const int m = v + 8 * khalf;
      sB[wave][m * TS + n] = gelu_erf(c[v] + bn);
    }
  }
  __syncthreads();

  // ---- stage 3: x_proc = sB @ W2 + b2 -> sA ----
  for (int nt = 0; nt < 4; ++nt) {
    v8f c = {};
    c = wmma_k(&sB[wave][0], TS, &sW2[nt * 16], DD, DD, c);
    const int n = nt * 16 + mrow;
    const float bn = sb2[n];
    for (int v = 0; v < 8; ++v) {
      const int m = v + 8 * khalf;
      sA[wave][m * TS + n] = c[v] + bn;
    }
  }
  __syncthreads();

  // ---- stage 4: Q = sA @ Wq -> sB ----
  for (int nt = 0; nt < 4; ++nt) {
    v8f c = {};
    c = wmma_k(&sA[wave][0], TS, &sWq[nt * 16], DD, DD, c);
    const int n = nt * 16 + mrow;
    for (int v = 0; v < 8; ++v) {
      const int m = v + 8 * khalf;
      sB[wave][m * TS + n] = c[v];
    }
  }
  __syncthreads();

  // ---- stage 5: attn = (Q @ Kb^T) / (sqrt(D)+eps) ----
  const float scale = 1.0f / (8.0f + 1e-8f);
  for (int q = 0; q < 4; ++q) {
    const int idx = lane * 4 + q;                  // 0..127
    const int row = idx >> 3, atom = idx & 7;
    float dot = 0.f;
    for (int d = 0; d < DD; ++d) dot += sB[wave][row * TS + d] * sKb[atom * DD + d];
    sAt[wave][row * AA + atom] = dot * scale;
  }
  __syncthreads();

  // ---- stage 6: softmax over A=8, entropy partial ----
  float entp = 0.f;
  if (lane < 16) {
    float a[AA];
    float mx = -3.4e38f;
    for (int k = 0; k < AA; ++k) { a[k] = sAt[wave][lane * AA + k]; mx = fmaxf(mx, a[k]); }
    float s = 0.f;
    for (int k = 0; k < AA; ++k) { a[k] = expf(a[k] - mx); s += a[k]; }
    const float inv = 1.0f / s;
    for (int k = 0; k < AA; ++k) {
      const float w = a[k] * inv;
      sAt[wave][lane * AA + k] = w;
      const float ws = w + 1e-8f;
      entp -= ws * logf(ws);
    }
  }
  for (int off = 16; off > 0; off >>= 1) entp += __shfl_down(entp, off);
  if (lane == 0) entpart[t] = entp;
  __syncthreads();

  // ---- stage 7: x_clean = clip(weights @ basis, +-5) -> global ----
  for (int it = 0; it < 32; ++it) {
    const int v = it * 32 + lane;
    const int row = v >> 6, d = v & 63;
    float acc = 0.f;
    for (int k = 0; k < AA; ++k) acc += sAt[wave][row * AA + k] * sBas[k * DD + d];
    acc = fminf(fmaxf(acc, -5.0f), 5.0f);
    xflat[(size_t)(r0 + row) * DD + d] = acc;
  }
}

// ---------------------------------------------------------------------------
// entropy reduction (deterministic fixed-order tree)
// ---------------------------------------------------------------------------
__global__ __launch_bounds__(256) void k_ent_reduce(const float* __restrict__ part,
                                                    float* __restrict__ out_ent) {
  __shared__ float s[256];
  float acc = 0.f;
  for (int i = threadIdx.x; i < NTILES; i += 256) acc += part[i];
  s[threadIdx.x] = acc;
  __syncthreads();
  for (int off = 128; off > 0; off >>= 1) {
    if (threadIdx.x < off) s[threadIdx.x] += s[threadIdx.x + off];
    __syncthreads();
  }
  if (threadIdx.x == 0) *out_ent = s[0] / (float)BN;
}

// ---------------------------------------------------------------------------
// K4a: split-K partials of x_flat @ Wr1   [512x65536]x[65536x128]
// wave computes 16(M) x 64(N) over a 2048-wide K chunk; 32 chunks.
// partial layout: [chunk][512][128]
// ---------------------------------------------------------------------------
__global__ __launch_bounds__(256) void k_gemm_readout1(const float* __restrict__ Xf,
                                                       const float* __restrict__ Wr1,
                                                       float* __restrict__ part) {
  const int wave = threadIdx.x >> 5;
  const int t = blockIdx.x * 8 + wave;            // 2048 wave tasks
  const int ch = t & (R1NCH - 1);                 // K chunk
  const int rem = t >> 5;
  const int nb = rem & 1;                         // 0..1 (64-wide)
  const int mt = rem >> 1;                        // 0..31
  const int m0 = mt * 16, n0 = nb * 64, k0 = ch * R1CHUNK;
  const int lane = threadIdx.x & 31;
  const int mrow = lane & 15, khalf = lane >> 4;

  const float* Arow = Xf + (size_t)(m0 + mrow) * ND;
  const float* Bp   = Wr1 + n0 + mrow;
  v8f c0 = {}, c1 = {}, c2 = {}, c3 = {};
  for (int k = k0; k < k0 + R1CHUNK; k += 4) {
    const int kk = k + 2 * khalf;
    v2f a, b;
    a.x = Arow[kk];
    a.y = Arow[kk + 1];
    const float* B0 = Bp + (size_t)kk * (2 * HH);
    const float* B1 = Bp + (size_t)(kk + 1) * (2 * HH);
    b.x = B0[0];  b.y = B1[0];  c0 = wmma4(a, b, c0);
    b.x = B0[16]; b.y = B1[16]; c1 = wmma4(a, b, c1);
    b.x = B0[32]; b.y = B1[32]; c2 = wmma4(a, b, c2);
    b.x = B0[48]; b.y = B1[48]; c3 = wmma4(a, b, c3);
  }
  v8f acc[4] = {c0, c1, c2, c3};
  float* pc = part + (size_t)ch * BATCH * (2 * HH);
  for (int j = 0; j < 4; ++j) {
    const int n = n0 + j * 16 + mrow;
    for (int v = 0; v < 8; ++v) {
      const int m = m0 + v + 8 * khalf;
      pc[(size_t)m * (2 * HH) + n] = acc[j][v];
    }
  }
}

// ---------------------------------------------------------------------------
// K4b: y1 = gelu(sum_ch partial + br1)   (fixed-order, deterministic)
// ---------------------------------------------------------------------------
__global__ __launch_bounds__(256) void k_r1_reduce(const float* __restrict__ part,
                                                   const float* __restrict__ br1,
                                                   float* __restrict__ Y1) {
  const int idx = blockIdx.x * 256 + threadIdx.x;  // < 512*128
  const int n = idx & (2 * HH - 1);
  float acc = 0.f;
  for (int ch = 0; ch < R1NCH; ++ch) acc += part[(size_t)ch * BATCH * (2 * HH) + idx];
  Y1[idx] = gelu_erf(acc + br1[n]);
}

// ---------------------------------------------------------------------------
// K5: logits = y1 @ Wr2 + br2   [512x128]x[128x100]
// ---------------------------------------------------------------------------
__global__ __launch_bounds__(256) void k_logits(const float* __restrict__ Y1,
                                                const float* __restrict__ Wr2,
                                                const float* __restrict__ br2,
                                                float* __restrict__ out) {
  const int idx = blockIdx.x * 256 + threadIdx.x;
  if (idx >= BATCH * CC) return;
  const int b = idx / CC, c = idx % CC;
  float acc = br2[c];
  for (int h = 0; h < 2 * HH; ++h) acc += Y1[b * (2 * HH) + h] * Wr2[h * CC + c];
  out[idx] = acc;
}

// ---------------------------------------------------------------------------
extern "C" void kernel_launch(void* const* d_in, const int* in_sizes, int n_in,
                              void* d_out, int out_size, void* d_ws, size_t ws_size,
                              hipStream_t stream) {
  (void)in_sizes; (void)n_in; (void)out_size; (void)ws_size;
  const float* x          = (const float*)d_in[0];
  const float* W_embed    = (const float*)d_in[1];
  const float* b_embed    = (const float*)d_in[2];
  const float* adj_w      = (const float*)d_in[3];
  const float* adj_m      = (const float*)d_in[4];
  const float* W1         = (const float*)d_in[5];
  const float* b1         = (const float*)d_in[6];
  const float* W2         = (const float*)d_in[7];
  const float* b2         = (const float*)d_in[8];
  const float* basis      = (const float*)d_in[9];
  const float* Wq         = (const float*)d_in[10];
  const float* Wk         = (const float*)d_in[11];
  const float* Wr1        = (const float*)d_in[12];
  const float* br1        = (const float*)d_in[13];
  const float* Wr2        = (const float*)d_in[14];
  const float* br2        = (const float*)d_in[15];

  float* out = (float*)d_out;                  // [51200 logits][1 entropy][1 ortho]
  float* w = (float*)d_ws;
  float* xemb   = w;                               // 33,554,432 f
  float* xflat  = xemb  + (size_t)BATCH * ND;      // 33,554,432 f
  float* y1     = xflat + (size_t)BATCH * ND;      //     65,536 f
  float* adj4   = y1    + (size_t)BATCH * 2 * HH;  //      4,096 f
  float* kb     = adj4  + (size_t)NN * 4;          //        512 f
  float* entp   = kb    + (size_t)AA * DD;         //     32,768 f
  float* r1part = entp  + (size_t)NTILES;          //  2,097,152 f

  k_prep<<<1, 256, 0, stream>>>(basis, Wk, kb, out + BATCH * CC + 1);
  k_adj<<<NN / 8, 256, 0, stream>>>(adj_w, adj_m, adj4);
  k_gemm_embed<<<(BATCH / 16) * (ND / 64) / 8, 256, 0, stream>>>(x, W_embed, b_embed, xemb);
  k_node<<<NTILES / 8, 256, 0, stream>>>(xemb, adj4, W1, b1, W2, b2, Wq, kb, basis,
                                         xflat, entp);
  k_ent_reduce<<<1, 256, 0, stream>>>(entp, out + BATCH * CC);
  k_gemm_readout1<<<(BATCH / 16) * (2 * HH / 64) * R1NCH / 8, 256, 0, stream>>>(xflat, Wr1, r1part);
  k_r1_reduce<<<BATCH * 2 * HH / 256, 256, 0, stream>>>(r1part, br1, y1);
  k_logits<<<(BATCH * CC + 255) / 256, 256, 0, stream>>>(y1, Wr2, br2, out);
}